// DDF_Conv2D_33079838113932
// MI455X (gfx1250) — compile-verified
//
#include <hip/hip_runtime.h>
#include <math.h>

typedef __attribute__((ext_vector_type(2))) float v2f;
typedef __attribute__((ext_vector_type(8))) float v8f;

#define BATCH 4
#define CH    128
#define HW    128
#define NPIX  (HW * HW)       // 16384
#define MID   25
#define KK    9
#define CF_N  (CH * KK)       // 1152

__device__ __forceinline__ float sigmoidf(float v) {
    return 1.0f / (1.0f + expf(-v));
}

// ---------------------------------------------------------------------------
// Kernel 1: per-(b,c) global average pool.  512 blocks x 256 threads.
// ---------------------------------------------------------------------------
__global__ void ddf_means(const float* __restrict__ x, float* __restrict__ xm) {
    int bc = blockIdx.x;                       // 0..511 = b*128 + c
    const float4* xv = (const float4*)(x + (size_t)bc * NPIX);
    int tid = threadIdx.x;
    float s = 0.f;
    #pragma unroll
    for (int i = tid; i < NPIX / 4; i += 256) {
        float4 v = xv[i];
        s += (v.x + v.y) + (v.z + v.w);
    }
    __shared__ float red[256];
    red[tid] = s;
    __syncthreads();
    for (int off = 128; off > 0; off >>= 1) {
        if (tid < off) red[tid] += red[tid + off];
        __syncthreads();
    }
    if (tid == 0) xm[bc] = red[0] * (1.0f / NPIX);
}

// ---------------------------------------------------------------------------
// Kernel 2: channel branch: hid = relu(xm@W1^T+b1); cf = sigmoid(hid@W2^T+b2)
// Tiny (130K MACs) -> one block of 256 threads, plain VALU.
// ---------------------------------------------------------------------------
__global__ void ddf_chfilter(const float* __restrict__ xm,
                             const float* __restrict__ W1,
                             const float* __restrict__ b1,
                             const float* __restrict__ W2,
                             const float* __restrict__ b2,
                             float* __restrict__ cf) {
    __shared__ float xms[BATCH][CH];
    __shared__ float hid[BATCH][MID];
    int tid = threadIdx.x;
    for (int i = tid; i < BATCH * CH; i += 256) ((float*)xms)[i] = xm[i];
    __syncthreads();
    if (tid < BATCH * MID) {
        int b = tid / MID, m = tid % MID;
        float acc = b1[m];
        for (int c = 0; c < CH; ++c) acc += xms[b][c] * W1[m * CH + c];
        hid[b][m] = fmaxf(acc, 0.0f);
    }
    __syncthreads();
    for (int idx = tid; idx < BATCH * CF_N; idx += 256) {
        int b = idx / CF_N, o = idx % CF_N;
        float acc = b2[o];
        #pragma unroll
        for (int m = 0; m < MID; ++m) acc += hid[b][m] * W2[o * MID + m];
        cf[idx] = sigmoidf(acc);
    }
}

// ---------------------------------------------------------------------------
// Kernel 3: spatial branch as WMMA GEMM.  Per batch: S[9,16384] =
// Ws[9,128] @ x[128,16384], then +bs and sigmoid.  M padded 9->16.
// One wave per 16x16 output tile: 32 x V_WMMA_F32_16X16X4_F32 over K=128.
// Padded A (16x128 Ws) is staged in LDS once per block (8 KB, shared by all
// 8 waves); each K-step fetches A with one aligned ds_load_b64 -> no VGPR
// array, no scratch spills.  Inner loop: ds_b64 + 2 coalesced global_b32 +
// WMMA.
// ---------------------------------------------------------------------------
__global__ void ddf_spatial_wmma(const float* __restrict__ x,
                                 const float* __restrict__ Ws,
                                 const float* __restrict__ bs,
                                 float* __restrict__ sfilt) {
    __shared__ float wsLds[16 * CH];                  // padded A, row-major [m][k]
    int tid = threadIdx.x;
    for (int i = tid; i < 16 * CH; i += 256) {
        int m = i >> 7, k = i & 127;
        wsLds[i] = (m < KK) ? Ws[m * CH + k] : 0.0f;
    }
    __syncthreads();

    int wave = blockIdx.x * 8 + (tid >> 5);           // 0..4095
    int lane = tid & 31;
    int b    = wave >> 10;                            // batch
    int base = (wave & 1023) << 4;                    // pixel tile start
    int col  = lane & 15;                             // M for A, N for B
    int kOff = (lane >> 4) << 1;                      // 0 or 2

    // A fetch base in LDS: row col, K offset kOff (8-byte aligned)
    const float* aRow = &wsLds[col * CH + kOff];
    // B fetch base: x column [K x 16 pixels]
    const float* xcol = x + (size_t)b * CH * NPIX + (size_t)kOff * NPIX
                          + base + col;

    v8f acc = {};
    #pragma unroll 8
    for (int kk = 0; kk < 32; ++kk) {
        v2f a = *(const v2f*)(aRow + kk * 4);         // ds_load_b64
        v2f bv;
        bv.x = xcol[(size_t)(kk * 4    ) * NPIX];
        bv.y = xcol[(size_t)(kk * 4 + 1) * NPIX];
        acc = __builtin_amdgcn_wmma_f32_16x16x4_f32(
            /*neg_a=*/false, a, /*neg_b=*/false, bv,
            /*c_mod=*/(short)0, acc, /*reuse_a=*/false, /*reuse_b=*/false);
    }

    // D layout: lanes 0-15 vgpr r = D[M=r][N=lane]; lanes 16-31 = D[M=r+8][N=lane-16]
    float* sb = sfilt + (size_t)b * KK * NPIX;
    if (lane < 16) {
        #pragma unroll
        for (int r = 0; r < 8; ++r)
            sb[r * NPIX + base + col] = sigmoidf(acc[r] + bs[r]);
    } else {
        sb[8 * NPIX + base + col] = sigmoidf(acc[0] + bs[8]);
    }
}

// ---------------------------------------------------------------------------
// Kernel 4: fused unfold * (spatial x channel filter) -> sum over 9 taps.
// Block = (b, h): caches the 9 spatial-filter values per pixel of this row in
// LDS once, then sweeps channels two at a time with x row-tiles (with halo)
// staged through LDS.  Never materializes the [b,c,9,h,w] tensor.
// ---------------------------------------------------------------------------
__global__ void ddf_main(const float* __restrict__ x,
                         const float* __restrict__ sfilt,
                         const float* __restrict__ cf,
                         float* __restrict__ out) {
    int bh = blockIdx.x;            // 0..511
    int b = bh >> 7;
    int h = bh & 127;
    int tid = threadIdx.x;
    int g = tid >> 7;               // which channel of the pair
    int w = tid & 127;

    __shared__ float ss[KK][HW];
    __shared__ float xs[2][3][HW + 4];   // 3 rows + column halo

    const float* sb = sfilt + (size_t)b * KK * NPIX + h * HW;
    for (int i = tid; i < KK * HW; i += 256)
        ss[i / HW][i % HW] = sb[(i / HW) * NPIX + (i % HW)];
    if (tid < 6) {                  // column halos are the zero padding
        int gg = tid & 1, r = tid >> 1;
        xs[gg][r][0] = 0.0f;
        xs[gg][r][HW + 1] = 0.0f;
    }
    __syncthreads();

    const float* xbase = x   + (size_t)b * CH * NPIX;
    float*       obase = out + (size_t)b * CH * NPIX;
    const float* cfb   = cf  + (size_t)b * CF_N;

    for (int cc = 0; cc < CH / 2; ++cc) {
        int c = cc * 2 + g;
        const float* xc = xbase + (size_t)c * NPIX;
        #pragma unroll
        for (int r = 0; r < 3; ++r) {
            int hh = h + r - 1;
            xs[g][r][w + 1] = (hh >= 0 && hh < HW) ? xc[hh * HW + w] : 0.0f;
        }
        __syncthreads();

        const float* cfc = cfb + c * KK;
        float cfv[KK];
        #pragma unroll
        for (int kk = 0; kk < KK; ++kk) cfv[kk] = cfc[kk];

        float o = 0.f;
        #pragma unroll
        for (int i = 0; i < 3; ++i)
            #pragma unroll
            for (int j = 0; j < 3; ++j)
                o += xs[g][i][w + j] * ss[i * 3 + j][w] * cfv[i * 3 + j];

        obase[(size_t)c * NPIX + h * HW + w] = o;
        __syncthreads();
    }
}

// ---------------------------------------------------------------------------
extern "C" void kernel_launch(void* const* d_in, const int* in_sizes, int n_in,
                              void* d_out, int out_size, void* d_ws, size_t ws_size,
                              hipStream_t stream) {
    const float* x  = (const float*)d_in[0];
    const float* Ws = (const float*)d_in[1];
    const float* bs = (const float*)d_in[2];
    const float* W1 = (const float*)d_in[3];
    const float* b1 = (const float*)d_in[4];
    const float* W2 = (const float*)d_in[5];
    const float* b2 = (const float*)d_in[6];
    float* out = (float*)d_out;

    // workspace layout: sfilt [4*9*16384] | xm [512] | cf [4608]  (~2.4 MB)
    float* sfilt = (float*)d_ws;
    float* xm    = sfilt + (size_t)BATCH * KK * NPIX;
    float* cfb   = xm + BATCH * CH;

    ddf_means       <<<BATCH * CH, 256, 0, stream>>>(x, xm);
    ddf_chfilter    <<<1,          256, 0, stream>>>(xm, W1, b1, W2, b2, cfb);
    ddf_spatial_wmma<<<512,        256, 0, stream>>>(x, Ws, bs, sfilt);
    ddf_main        <<<512,        256, 0, stream>>>(x, sfilt, cfb, out);
}